// LightGCNMultiAtt_69733089018091
// MI455X (gfx1250) — compile-verified
//
#include <hip/hip_runtime.h>

#define NUM_USERS  100000
#define NUM_ITEMS  50000
#define NUM_OTHERS 20000
#define N_NODES    (NUM_USERS + NUM_ITEMS + NUM_OTHERS)
#define DIM        64
#define NNZ_TOT    5000000
#define BATCH_SZ   8192
#define N_LAYERS   3

typedef __attribute__((ext_vector_type(2))) float v2f;
typedef __attribute__((ext_vector_type(8))) float v8f;

// ---------------------------------------------------------------------------
// 1) init: cur = acc = concat(user_emb, item_emb, other_emb)   (float4 vec)
// ---------------------------------------------------------------------------
__global__ void lgn_init(const float4* __restrict__ ue,
                         const float4* __restrict__ ie,
                         const float4* __restrict__ oe,
                         float4* __restrict__ cur,
                         float4* __restrict__ acc) {
  int i = blockIdx.x * blockDim.x + threadIdx.x;
  if (i >= N_NODES * (DIM / 4)) return;
  int node = i >> 4;          // 16 float4 per 64-dim row
  int q    = i & 15;
  float4 v;
  if (node < NUM_USERS)                   v = ue[node * 16 + q];
  else if (node < NUM_USERS + NUM_ITEMS)  v = ie[(node - NUM_USERS) * 16 + q];
  else                                    v = oe[(node - NUM_USERS - NUM_ITEMS) * 16 + q];
  cur[i] = v;
  acc[i] = v;
}

// ---------------------------------------------------------------------------
// 2) zero a node-buffer (float4 vec)
// ---------------------------------------------------------------------------
__global__ void lgn_zero(float4* __restrict__ p) {
  int i = blockIdx.x * blockDim.x + threadIdx.x;
  if (i >= N_NODES * (DIM / 4)) return;
  p[i] = make_float4(0.f, 0.f, 0.f, 0.f);
}

// ---------------------------------------------------------------------------
// 3) SpMM scatter: y[row[e]] += vals[e] * x[col[e]]
//    One wave32 handles 32 consecutive edges: coalesced metadata load, then
//    broadcast each edge via __shfl; lanes cover 64 dims as float2 gathers
//    (L2-resident, table fits in 192MB L2) + 2 fp32 atomics per lane.
// ---------------------------------------------------------------------------
__global__ void lgn_spmm(const float* __restrict__ x,
                         const float* __restrict__ vals,
                         const int*   __restrict__ row,
                         const int*   __restrict__ col,
                         float* __restrict__ y, int nnz) {
  int wave = (blockIdx.x * blockDim.x + threadIdx.x) >> 5;
  int lane = threadIdx.x & 31;
  int base = wave * 32;
  if (base >= nnz) return;

  int   r = 0, c = 0;
  float v = 0.f;
  int e = base + lane;
  if (e < nnz) { r = row[e]; c = col[e]; v = vals[e]; }

  int cnt = nnz - base; if (cnt > 32) cnt = 32;
  for (int j = 0; j < cnt; ++j) {
    int   rj = __shfl(r, j);
    int   cj = __shfl(c, j);
    float vj = __shfl(v, j);
    const float2 xv = *(const float2*)(x + cj * DIM + lane * 2);
    float* dst = y + rj * DIM + lane * 2;
    atomicAdd(dst,     vj * xv.x);
    atomicAdd(dst + 1, vj * xv.y);
  }
}

// ---------------------------------------------------------------------------
// 4) acc += nxt (float4 vec)
// ---------------------------------------------------------------------------
__global__ void lgn_accum(float4* __restrict__ acc, const float4* __restrict__ nxt) {
  int i = blockIdx.x * blockDim.x + threadIdx.x;
  if (i >= N_NODES * (DIM / 4)) return;
  float4 a = acc[i];
  float4 b = nxt[i];
  a.x += b.x; a.y += b.y; a.z += b.z; a.w += b.w;
  acc[i] = a;
}

// ---------------------------------------------------------------------------
// 5) gamma via V_WMMA_F32_16X16X4_F32:
//    One wave per 16 (user,item) pairs. A = U(16x64), B = I^T(64x16),
//    D = A x B accumulated over 16 WMMA steps (K=4 each); diagonal of D
//    holds the 16 dot products. Scale 1/16 folds the two acc/4 factors.
//    32-bit A 16x4 layout (ISA 7.12.2): lane L -> M = L&15, VGPR{0,1} hold
//    K = 2*(L>>4) + {0,1}. B 4x16: lane L -> N = L&15, VGPR{0,1} hold
//    K = 2*(L>>4) + {0,1}. D 16x16: VGPR r, lane L -> M = r + 8*(L>>4),
//    N = L&15.
// ---------------------------------------------------------------------------
__global__ void lgn_gamma_wmma(const float* __restrict__ acc,
                               const int*   __restrict__ users,
                               const int*   __restrict__ items,
                               float* __restrict__ out, int batch) {
  int wave = (blockIdx.x * blockDim.x + threadIdx.x) >> 5;
  int lane = threadIdx.x & 31;
  int p0 = wave * 16;
  if (p0 >= batch) return;           // whole-wave exit: EXEC stays all-ones

  int g = lane >> 4;                 // lane half selects K sub-pair
  int n = lane & 15;                 // row (A) / col (B) index

  int u  = users[p0 + n];
  int it = items[p0 + n];
  const float* ua = acc + u * DIM;
  const float* ia = acc + (NUM_USERS + it) * DIM;

  v8f d = {};
  #pragma unroll
  for (int k = 0; k < DIM / 4; ++k) {
    int off = k * 4 + g * 2;
    v2f a, b;
    a.x = ua[off]; a.y = ua[off + 1];
    b.x = ia[off]; b.y = ia[off + 1];
    d = __builtin_amdgcn_wmma_f32_16x16x4_f32(
        /*neg_a=*/false, a, /*neg_b=*/false, b,
        /*c_mod=*/(short)0, d, /*reuse_a=*/false, /*reuse_b=*/false);
  }

  // Extract diagonal element p==n: M = r + 8*g must equal n.
  if (g == 0 && n < 8)  out[p0 + n] = d[n]     * (1.0f / 16.0f);
  if (g == 1 && n >= 8) out[p0 + n] = d[n - 8] * (1.0f / 16.0f);
}

// ---------------------------------------------------------------------------
// host-side launch
// ---------------------------------------------------------------------------
extern "C" void kernel_launch(void* const* d_in, const int* in_sizes, int n_in,
                              void* d_out, int out_size, void* d_ws, size_t ws_size,
                              hipStream_t stream) {
  const float* user_emb  = (const float*)d_in[0];
  const float* item_emb  = (const float*)d_in[1];
  const float* other_emb = (const float*)d_in[2];
  const float* vals      = (const float*)d_in[3];
  const int*   row_idx   = (const int*)  d_in[4];
  const int*   col_idx   = (const int*)  d_in[5];
  const int*   users     = (const int*)  d_in[6];
  const int*   items     = (const int*)  d_in[7];
  float*       out       = (float*)d_out;

  const size_t rowElems = (size_t)N_NODES * DIM;
  float* cur = (float*)d_ws;
  float* nxt = cur + rowElems;
  float* acc = nxt + rowElems;

  const int vec4  = N_NODES * (DIM / 4);
  const int bEl   = 256;
  const int gEl   = (vec4 + bEl - 1) / bEl;

  lgn_init<<<gEl, bEl, 0, stream>>>((const float4*)user_emb, (const float4*)item_emb,
                                    (const float4*)other_emb, (float4*)cur, (float4*)acc);

  const int nnz     = in_sizes[3];
  const int waves   = (nnz + 31) / 32;
  const int gSpmm   = (waves * 32 + bEl - 1) / bEl;

  for (int l = 0; l < N_LAYERS; ++l) {
    lgn_zero<<<gEl, bEl, 0, stream>>>((float4*)nxt);
    lgn_spmm<<<gSpmm, bEl, 0, stream>>>(cur, vals, row_idx, col_idx, nxt, nnz);
    lgn_accum<<<gEl, bEl, 0, stream>>>((float4*)acc, (const float4*)nxt);
    float* t = cur; cur = nxt; nxt = t;   // resolved at capture time; deterministic
  }

  const int tiles  = (BATCH_SZ + 15) / 16;         // 512 waves
  const int gGamma = (tiles * 32 + bEl - 1) / bEl; // 64 blocks
  lgn_gamma_wmma<<<gGamma, bEl, 0, stream>>>(acc, users, items, out, BATCH_SZ);
}